// SNN_59176059404494
// MI455X (gfx1250) — compile-verified
//
#include <hip/hip_runtime.h>
#include <cstddef>
#include <cstdint>

// CDNA5 / gfx1250 fused SNN forward, async-LDS pipelined.
//
// Per wave: owns one batch row b. x-tiles (96 rows x 16 t, 6 KB) are DMA'd
// into double-buffered LDS with GLOBAL_LOAD_ASYNC_TO_LDS_B64 one chunk ahead
// (ASYNCcnt), hiding HBM latency with no B-fragment register staging.
// GEMM: cur1[h][t] = sum_i W1[h][i]*x[b][i][t] via V_WMMA_F32_16X16X4_F32.
// W1 A-fragments are explicitly hoisted into ~192 VGPRs once per wave (SSA
// values are immune to the asm memory clobbers, unlike LDS re-reads).
// C-fragments transpose through curT[16][68], then a serial LIF scan with
// register-resident membranes and a __shfl_xor butterfly for the 50->2 layer.

typedef float v2f __attribute__((ext_vector_type(2)));
typedef float v4f __attribute__((ext_vector_type(4)));
typedef float v8f __attribute__((ext_vector_type(8)));

#define BETA 0.96f
#define THRESH 1.0f

constexpr int I    = 96;        // inputs (K)
constexpr int H    = 50;        // hidden neurons
constexpr int HP   = 64;        // padded hidden (4 WMMA M-tiles)
constexpr int T    = 250;       // time steps
constexpr int CT   = 16;        // t-chunk = WMMA N
constexpr int KS   = I / 4;     // 24 K-steps per chunk
constexpr int LROW = HP + 4;    // 68: 16B-aligned rows, conflict-light banks
constexpr int NC   = 16;        // chunks (last covers t 240..249)
constexpr int XT_F = I * CT;    // 1536 floats per x tile (6 KB)
constexpr int W1_F = HP * I;    // 6144 floats
constexpr int LDS_FLOATS = W1_F + 4 * 2 * XT_F + 4 * CT * LROW;  // 22784

// full chunk: 96 rows x 64 B, as 8-byte segments (row stride 1000 B => 8B align)
__device__ __forceinline__ void async_tile_full(unsigned ldsb, unsigned long long gb, int lane) {
    const int r0 = lane >> 3;          // 0..3
    const int sg = lane & 7;           // 8B segment within 64B row
#pragma unroll
    for (int s = 0; s < 24; ++s) {     // rows s*4 + r0
        const unsigned           ld = ldsb + (unsigned)((s * 4 + r0) * 64 + sg * 8);
        const unsigned long long ga = gb + (unsigned long long)(s * 4 + r0) * (T * 4) + sg * 8;
        asm volatile("global_load_async_to_lds_b64 %0, %1, off" :: "v"(ld), "v"(ga) : "memory");
    }
}

// tail chunk (t0=240): only t 240..249 exist -> 5 in-bounds 8B segments per row
__device__ __forceinline__ void async_tile_tail(unsigned ldsb, unsigned long long gb, int lane) {
    const int r0 = lane >> 2;          // 0..7
    const int sg = lane & 3;           // segments t 240..247
#pragma unroll
    for (int s = 0; s < 12; ++s) {     // rows s*8 + r0
        const unsigned           ld = ldsb + (unsigned)((s * 8 + r0) * 64 + sg * 8);
        const unsigned long long ga = gb + (unsigned long long)(s * 8 + r0) * (T * 4) + sg * 8;
        asm volatile("global_load_async_to_lds_b64 %0, %1, off" :: "v"(ld), "v"(ga) : "memory");
    }
#pragma unroll
    for (int s = 0; s < 3; ++s) {      // segment t 248..249 (bytes 992..999)
        const int r = s * 32 + lane;
        const unsigned           ld = ldsb + (unsigned)(r * 64 + 32);
        const unsigned long long ga = gb + (unsigned long long)r * (T * 4) + 32;
        asm volatile("global_load_async_to_lds_b64 %0, %1, off" :: "v"(ld), "v"(ga) : "memory");
    }
}

__global__ __launch_bounds__(128)
__attribute__((amdgpu_num_vgpr(256)))
void snn_fused_kernel(const float* __restrict__ x,
                      const float* __restrict__ W1,
                      const float* __restrict__ W2,
                      float* __restrict__ out) {
    extern __shared__ float ldsbuf[];
    float* const W1s = ldsbuf;                                   // [64][96]

    const int tid  = threadIdx.x;
    const int lane = tid & 31;
    const int wv   = tid >> 5;
    const int b    = blockIdx.x * 4 + wv;        // one batch row per wave

    float* const xt0 = ldsbuf + W1_F + wv * 2 * XT_F;            // ping
    float* const xt1 = xt0 + XT_F;                               // pong
    float* const myT = ldsbuf + W1_F + 8 * XT_F + wv * (CT * LROW);

    // ---- stage W1 into LDS, zero-padded to 64 rows
    for (int idx = tid; idx < W1_F; idx += 128) {
        const int h = idx / I, i = idx % I;
        W1s[idx] = (h < H) ? W1[h * I + i] : 0.0f;
    }
    __syncthreads();

    const unsigned long long xg  = (unsigned long long)(const void*)(x + (size_t)b * I * T);
    const unsigned           xl0 = (unsigned)(uintptr_t)(void*)xt0;  // LDS byte offset
    const unsigned           xl1 = (unsigned)(uintptr_t)(void*)xt1;

    // ---- prologue: DMA chunk 0 into ping
    async_tile_full(xl0, xg, lane);

    // ---- per-lane W2 coefficients for h = lane and h = lane + 32
    const float w2a0 = W2[lane];
    const float w2a1 = W2[H + lane];
    const float w2b0 = (lane + 32 < H) ? W2[lane + 32]     : 0.0f;
    const float w2b1 = (lane + 32 < H) ? W2[H + lane + 32] : 0.0f;

    const int n  = lane & 15;   // WMMA: A-frag row-in-tile (h), B/C-frag col (t)
    const int hg = lane >> 4;   // half-wave group -> K pair select

    // ---- hoist ALL W1 A-fragments into registers (once per wave, ~192 VGPRs)
    const float* const w1p = &W1s[n * I + 2 * hg];
    v2f areg[KS][4];
#pragma unroll
    for (int ks = 0; ks < KS; ++ks) {
#pragma unroll
        for (int mb = 0; mb < 4; ++mb) {
            areg[ks][mb] = *(const v2f*)(w1p + mb * 16 * I + ks * 4);
        }
    }

    float mem1a = 0.0f, mem1b = 0.0f;   // lif1 membrane, register-resident
    float mem2_0 = 0.0f, mem2_1 = 0.0f; // lif2 membrane (replicated per lane)

    float* const outb = out + (size_t)b * 2 * T;

    for (int ci = 0; ci < NC; ++ci) {
        // ---- DMA next chunk into the other buffer, then wait for current
        if (ci + 1 < NC) {
            const unsigned           nl = ((ci + 1) & 1) ? xl1 : xl0;
            const unsigned long long ng = xg + (unsigned)((ci + 1) * CT * 4);
            if (ci + 1 == NC - 1) {
                async_tile_tail(nl, ng, lane);
                asm volatile("s_wait_asynccnt 0xf" ::: "memory");   // 15 in flight
            } else {
                async_tile_full(nl, ng, lane);
                asm volatile("s_wait_asynccnt 0x18" ::: "memory");  // 24 in flight
            }
        } else {
            asm volatile("s_wait_asynccnt 0x0" ::: "memory");
        }

        // ---- GEMM from the LDS tile: B-frag = tile[4*ks+2*hg .. +1][n]
        const float* const bp = ((ci & 1) ? xt1 : xt0) + 32 * hg + n;
        v8f acc[4] = {v8f{0.f}, v8f{0.f}, v8f{0.f}, v8f{0.f}};
#pragma unroll
        for (int ks = 0; ks < KS; ++ks) {
            v2f bf;
            bf.x = bp[ks * 64];        // x[b][4ks+2hg  ][t0+n]
            bf.y = bp[ks * 64 + 16];   // x[b][4ks+2hg+1][t0+n]
#pragma unroll
            for (int mb = 0; mb < 4; ++mb) {
                acc[mb] = __builtin_amdgcn_wmma_f32_16x16x4_f32(
                    false, areg[ks][mb], false, bf, (short)0, acc[mb], false, false);
            }
        }

        // ---- transpose C-frags (m=h, n=t) into LDS as [t][h], b128 stores
#pragma unroll
        for (int mb = 0; mb < 4; ++mb) {
            const int base = n * LROW + mb * 16 + 8 * hg;
            *(v4f*)&myT[base]     = v4f{acc[mb][0], acc[mb][1], acc[mb][2], acc[mb][3]};
            *(v4f*)&myT[base + 4] = v4f{acc[mb][4], acc[mb][5], acc[mb][6], acc[mb][7]};
        }

        // ---- serial LIF scan over this chunk (same-wave LDS, no barrier)
        const int t0   = ci * CT;
        const int tmax = (T - t0 < CT) ? (T - t0) : CT;
        for (int tl = 0; tl < tmax; ++tl) {
            const float ca = myT[tl * LROW + lane];        // cur1, h = lane
            const float cb = myT[tl * LROW + lane + 32];   // cur1, h = lane+32

            // lif1: reset-to-zero uses OLD membrane vs threshold
            const float ba = BETA * mem1a + ca;
            const float bb = BETA * mem1b + cb;
            mem1a = (mem1a > THRESH) ? 0.0f : ba;
            mem1b = (mem1b > THRESH) ? 0.0f : bb;

            // spike = (mem_new > THRESH); fold W2 weight in via select
            float p0 = ((mem1a > THRESH) ? w2a0 : 0.0f)
                     + ((mem1b > THRESH) ? w2b0 : 0.0f);
            float p1 = ((mem1a > THRESH) ? w2a1 : 0.0f)
                     + ((mem1b > THRESH) ? w2b1 : 0.0f);

            // cur2 = sum over 50 h's -> full-wave butterfly reduction
#pragma unroll
            for (int m = 16; m >= 1; m >>= 1) {
                p0 += __shfl_xor(p0, m, 32);
                p1 += __shfl_xor(p1, m, 32);
            }

            mem2_0 = BETA * mem2_0 + p0;   // lif2: no reset, record membrane
            mem2_1 = BETA * mem2_1 + p1;

            if (lane < 2) {
                outb[(size_t)lane * T + (t0 + tl)] = (lane == 0) ? mem2_0 : mem2_1;
            }
        }
    }
}

extern "C" void kernel_launch(void* const* d_in, const int* in_sizes, int n_in,
                              void* d_out, int out_size, void* d_ws, size_t ws_size,
                              hipStream_t stream) {
    (void)n_in; (void)out_size; (void)d_ws; (void)ws_size;
    const float* x  = (const float*)d_in[0];   // [B, 96, 250] f32
    const float* W1 = (const float*)d_in[1];   // [50, 96]     f32
    const float* W2 = (const float*)d_in[2];   // [2, 50]      f32
    float* out = (float*)d_out;                // [B, 2, 250]  f32

    const int B = in_sizes[0] / (96 * 250);    // 4096
    dim3 grid(B / 4);                          // 4 batch rows (waves) per block
    dim3 block(128);                           // 4 wave32
    hipLaunchKernelGGL(snn_fused_kernel, grid, block, LDS_FLOATS * 4, stream,
                       x, W1, W2, out);
}